// GCNEncoder_1975684956785
// MI455X (gfx1250) — compile-verified
//
#include <hip/hip_runtime.h>

typedef __attribute__((ext_vector_type(2))) float v2f;
typedef __attribute__((ext_vector_type(8))) float v8f;

#define FDIM 128

__device__ __forceinline__ void atomic_add_f32(float* p, float v) {
    // relaxed, agent scope, no return -> global_atomic_add_f32 (STOREcnt path)
    __hip_atomic_fetch_add(p, v, __ATOMIC_RELAXED, __HIP_MEMORY_SCOPE_AGENT);
}

// ---------------------------------------------------------------------------
// deg[v] = 1.0 (self loop)
__global__ __launch_bounds__(256) void gcn_deg_init(float* deg, int n) {
    int i = blockIdx.x * blockDim.x + threadIdx.x;
    if (i < n) deg[i] = 1.0f;
}

// deg[dst[e]] += 1
__global__ __launch_bounds__(256) void gcn_deg_count(const int* __restrict__ dst,
                                                     float* deg, int e) {
    int i = blockIdx.x * blockDim.x + threadIdx.x;
    if (i < e) atomic_add_f32(&deg[dst[i]], 1.0f);
}

// dinv[v] = rsqrt(deg[v])   (deg >= 1 always, no zero branch needed)
__global__ __launch_bounds__(256) void gcn_dinv(const float* __restrict__ deg,
                                                float* __restrict__ dinv, int n) {
    int i = blockIdx.x * blockDim.x + threadIdx.x;
    if (i < n) dinv[i] = rsqrtf(deg[i]);
}

// ---------------------------------------------------------------------------
// H[M,128] = A[M,128] @ W[128,128], f32 WMMA 16x16x4, one wave per 16x16 tile.
// grid.x = (M/16) * (128/16), blockDim = 32 (one wave32).
__global__ __launch_bounds__(32) void gcn_gemm_wmma(const float* __restrict__ A,
                                                    const float* __restrict__ W,
                                                    float* __restrict__ H, int M) {
    const int tile = blockIdx.x;
    const int tn   = tile & 7;        // 128/16 = 8 tiles along N
    const int tm   = tile >> 3;
    const int m0   = tm * 16;
    const int n0   = tn * 16;

    const int lane = threadIdx.x;     // 0..31 (wave32)
    const int half = lane >> 4;       // 0: K={0,1}, 1: K={2,3} within each K4 group
    const int l15  = lane & 15;       // A: row M, B/C/D: col N
    const int kofs = half * 2;

    const float* __restrict__ Arow = A + (size_t)(m0 + l15) * FDIM;

    v8f acc = {};
#pragma unroll
    for (int k = 0; k < FDIM; k += 4) {
        // 32-bit A-matrix 16x4 layout (ISA 7.12.2): lanes 0-15 hold K=0,1; 16-31 hold K=2,3
        v2f a = *(const v2f*)(Arow + k + kofs);
        // B 4x16: lanes hold N; VGPR0/1 = consecutive K within the half
        v2f b;
        b.x = W[(size_t)(k + kofs)     * FDIM + n0 + l15];
        b.y = W[(size_t)(k + kofs + 1) * FDIM + n0 + l15];
        acc = __builtin_amdgcn_wmma_f32_16x16x4_f32(
            /*neg_a=*/false, a, /*neg_b=*/false, b,
            /*c_mod=*/(short)0, acc, /*reuse_a=*/false, /*reuse_b=*/false);
    }

    // 32-bit C/D 16x16 layout: VGPR r -> M = m0 + r + (lane>=16 ? 8 : 0), N = n0 + (lane&15)
    const int row_base = m0 + half * 8;
#pragma unroll
    for (int r = 0; r < 8; ++r) {
        H[(size_t)(row_base + r) * FDIM + n0 + l15] = acc[r];
    }
}

// ---------------------------------------------------------------------------
// out[v,:] = H[v,:] * dinv[v]^2      (self-loop term; also fully initializes out)
__global__ __launch_bounds__(256) void gcn_self_init(const float* __restrict__ H,
                                                     const float* __restrict__ dinv,
                                                     float* __restrict__ out, int n) {
    size_t idx = (size_t)blockIdx.x * blockDim.x + threadIdx.x;
    size_t total = (size_t)n * FDIM;
    if (idx < total) {
        int v = (int)(idx >> 7);
        float di = dinv[v];
        out[idx] = H[idx] * di * di;
    }
}

// out[dst[e],f] += H[src[e],f] * dinv[src]*dinv[dst]   for all 128 features
// block = 256 threads = 2 edge slots x 128 features; EPB edges per block.
#define EPB 16
__global__ __launch_bounds__(256) void gcn_scatter(const int* __restrict__ src,
                                                   const int* __restrict__ dst,
                                                   const float* __restrict__ H,
                                                   const float* __restrict__ dinv,
                                                   float* __restrict__ out, int E) {
    const int f   = threadIdx.x & (FDIM - 1);
    const int sub = threadIdx.x >> 7;                 // 0 or 1
    long long e0 = (long long)blockIdx.x * EPB + sub;
    long long e1 = (long long)(blockIdx.x + 1) * EPB;
    if (e1 > E) e1 = E;
    for (long long e = e0; e < e1; e += 2) {
        int s = src[e];
        int d = dst[e];
        float w = dinv[s] * dinv[d];
        float val = H[(size_t)s * FDIM + f] * w;
        atomic_add_f32(&out[(size_t)d * FDIM + f], val);
    }
}

// x = relu(x + b[f])   in place
__global__ __launch_bounds__(256) void gcn_bias_relu(float* __restrict__ x,
                                                     const float* __restrict__ b, int n) {
    size_t idx = (size_t)blockIdx.x * blockDim.x + threadIdx.x;
    size_t total = (size_t)n * FDIM;
    if (idx < total) {
        float v = x[idx] + b[idx & (FDIM - 1)];
        x[idx] = v > 0.0f ? v : 0.0f;
    }
}

// ---------------------------------------------------------------------------
extern "C" void kernel_launch(void* const* d_in, const int* in_sizes, int n_in,
                              void* d_out, int out_size, void* d_ws, size_t ws_size,
                              hipStream_t stream) {
    const float* x  = (const float*)d_in[0];
    const int*   ei = (const int*)d_in[1];   // edge_index [2,E] (int32 on device)
    const float* W1 = (const float*)d_in[2];
    const float* b1 = (const float*)d_in[3];
    const float* W2 = (const float*)d_in[4];
    const float* b2 = (const float*)d_in[5];
    float* out = (float*)d_out;

    const int N = in_sizes[0] / FDIM;        // 50000
    const int E = in_sizes[1] / 2;           // 625000
    const int* src = ei;
    const int* dst = ei + E;

    // Workspace layout
    char* ws = (char*)d_ws;
    const size_t feat_bytes = (size_t)N * FDIM * sizeof(float);   // 25.6 MB
    float* H    = (float*)(ws);                                   // GEMM result
    float* AGG  = (float*)(ws + feat_bytes);                      // layer-1 accumulator
    float* deg  = (float*)(ws + 2 * feat_bytes);
    float* dinv = (float*)(ws + 2 * feat_bytes + ((size_t)N * 4 + 255) / 256 * 256);

    const size_t total = (size_t)N * FDIM;
    const int gN   = (N + 255) / 256;
    const int gE   = (E + 255) / 256;
    const int gTot = (int)((total + 255) / 256);
    const int gGemm = (N / 16) * (FDIM / 16);       // N % 16 == 0 for this workload
    const int gScat = (E + EPB - 1) / EPB;

    // ---- normalization ----
    gcn_deg_init <<<gN, 256, 0, stream>>>(deg, N);
    gcn_deg_count<<<gE, 256, 0, stream>>>(dst, deg, E);
    gcn_dinv     <<<gN, 256, 0, stream>>>(deg, dinv, N);

    // ---- layer 1: H = x @ W1 ; AGG = norm-aggregate(H) ; AGG = relu(AGG + b1) ----
    gcn_gemm_wmma <<<gGemm, 32, 0, stream>>>(x, W1, H, N);
    gcn_self_init <<<gTot, 256, 0, stream>>>(H, dinv, AGG, N);
    gcn_scatter   <<<gScat, 256, 0, stream>>>(src, dst, H, dinv, AGG, E);
    gcn_bias_relu <<<gTot, 256, 0, stream>>>(AGG, b1, N);

    // ---- layer 2: H = AGG @ W2 ; out = norm-aggregate(H) ; out = relu(out + b2) ----
    gcn_gemm_wmma <<<gGemm, 32, 0, stream>>>(AGG, W2, H, N);
    gcn_self_init <<<gTot, 256, 0, stream>>>(H, dinv, out, N);
    gcn_scatter   <<<gScat, 256, 0, stream>>>(src, dst, H, dinv, out, E);
    gcn_bias_relu <<<gTot, 256, 0, stream>>>(out, b2, N);
}